// RCNNClassifier_2module_42030549958868
// MI455X (gfx1250) — compile-verified
//
#include <hip/hip_runtime.h>

// ---- CDNA5 (gfx1250) wave32 ----
typedef float v2f __attribute__((ext_vector_type(2)));
typedef float v8f __attribute__((ext_vector_type(8)));
typedef float f4  __attribute__((ext_vector_type(4)));

#define B_SZ 128
#define N_SZ 65536
#define H_SZ 30
#define LOSS_GRID 1024

// ---------------------------------------------------------------------------
// Kernel 1: streaming copy RCNN_cls_result -> d_out[0 .. B*N*2)
// 64 MB read + 64 MB write, each byte touched once -> nontemporal b128 ops.
// ---------------------------------------------------------------------------
__global__ void copy_rcnn_kernel(const f4* __restrict__ src,
                                 f4* __restrict__ dst, int n4) {
    int i = blockIdx.x * blockDim.x + threadIdx.x;
    int stride = gridDim.x * blockDim.x;
    for (; i < n4; i += stride) {
        f4 v = __builtin_nontemporal_load(&src[i]);
        __builtin_nontemporal_store(v, &dst[i]);
    }
}

// ---------------------------------------------------------------------------
// Kernel 2: gt_last[b] = sum_h y[b,h,:]  (B=128 pairs into workspace)
// ---------------------------------------------------------------------------
__global__ void gt_kernel(const float* __restrict__ y, float* __restrict__ gt) {
    int b = threadIdx.x;
    if (b < B_SZ) {
        const float* p = y + b * (H_SZ * 2);
        float gx = 0.0f, gy = 0.0f;
#pragma unroll
        for (int h = 0; h < H_SZ; ++h) {
            gx += p[2 * h];
            gy += p[2 * h + 1];
        }
        gt[2 * b]     = gx;
        gt[2 * b + 1] = gy;
    }
}

// ---------------------------------------------------------------------------
// Kernel 3: per-block partial of
//   loss = sum_{b,n} (gx_b - (cx+ox))^2 + (gy_b - (cy+oy))^2
// Grid-stride over float4 (= 2 points). 8,388,608 points / (1024*256 lanes)
// / 2 points = 16 iterations for every lane -> no divergence, EXEC all-ones
// at the WMMA (ISA requirement).
//
// Wave reduction trick: D = A(16x4) x ones(4x16) + 0.
// A layout (32-bit 16x4): VGPR0 lanes0-15 -> (M=lane,K=0), lanes16-31 ->
// (M=lane-16,K=2); VGPR1 -> K=1/K=3 (set to 0). So D[m,n] = acc[m]+acc[m+16].
// C/D layout: lane L<16 holds D[0..7, L] in c[0..7]; lane L>=16 holds
// D[8..15, L-16]. sum(c[0..7]) gives half-wave row sums; one ds_swizzle
// SWAPX16 (offset 0x401F) adds the two halves -> full wave32 sum in one
// matrix op + one swizzle.
// ---------------------------------------------------------------------------
__global__ void loss_partials_kernel(const f4* __restrict__ cand,
                                     const f4* __restrict__ off,
                                     const float* __restrict__ gt,
                                     float* __restrict__ partials, int nq) {
    __shared__ float sgt[2 * B_SZ];
    __shared__ float wsum[8];

    for (int i = threadIdx.x; i < 2 * B_SZ; i += blockDim.x) sgt[i] = gt[i];
    __syncthreads();

    int tid = blockIdx.x * blockDim.x + threadIdx.x;
    int stride = gridDim.x * blockDim.x;
    float acc = 0.0f;

    for (int q = tid; q < nq; q += stride) {
        f4 c = __builtin_nontemporal_load(&cand[q]);   // 2 points, read once
        f4 o = __builtin_nontemporal_load(&off[q]);
        int b = q >> 15;             // N*2/4 = 32768 f4 per batch; 2 pts share b
        float gx = sgt[2 * b], gy = sgt[2 * b + 1];
        float dx0 = gx - (c[0] + o[0]);
        float dy0 = gy - (c[1] + o[1]);
        float dx1 = gx - (c[2] + o[2]);
        float dy1 = gy - (c[3] + o[3]);
        acc = fmaf(dx0, dx0, acc);
        acc = fmaf(dy0, dy0, acc);
        acc = fmaf(dx1, dx1, acc);
        acc = fmaf(dy1, dy1, acc);
    }

    // ---- wave32 reduction via V_WMMA_F32_16X16X4_F32 ----
    v2f a;  a[0] = acc;  a[1] = 0.0f;      // A: K=0/K=2 carry lane accs
    v2f bo; bo[0] = 1.0f; bo[1] = 1.0f;    // B: all ones (layout-agnostic)
    v8f cm = {0.f, 0.f, 0.f, 0.f, 0.f, 0.f, 0.f, 0.f};
    cm = __builtin_amdgcn_wmma_f32_16x16x4_f32(false, a, false, bo,
                                               (short)0, cm, false, false);
    float s = 0.0f;
#pragma unroll
    for (int i = 0; i < 8; ++i) s += cm[i];
    // add the two half-wave row-sum groups: SWAPX16 (xor=0x10, and=0x1F)
    s += __int_as_float(__builtin_amdgcn_ds_swizzle(__float_as_int(s), 0x401F));

    int lane = threadIdx.x & 31;
    int wave = threadIdx.x >> 5;
    if (lane == 0) wsum[wave] = s;
    __syncthreads();
    if (threadIdx.x == 0) {
        float t = 0.0f;
#pragma unroll
        for (int w = 0; w < 8; ++w) t += wsum[w];
        partials[blockIdx.x] = t;
    }
}

// ---------------------------------------------------------------------------
// Kernel 4: deterministic fixed-order reduction of block partials -> loss
// ---------------------------------------------------------------------------
__global__ void final_reduce_kernel(const float* __restrict__ partials, int n,
                                    float* __restrict__ out) {
    __shared__ float red[256];
    float s = 0.0f;
    for (int i = threadIdx.x; i < n; i += blockDim.x) s += partials[i];
    red[threadIdx.x] = s;
    __syncthreads();
    for (int w = 128; w > 0; w >>= 1) {
        if ((int)threadIdx.x < w) red[threadIdx.x] += red[threadIdx.x + w];
        __syncthreads();
    }
    if (threadIdx.x == 0) out[0] = red[0];
}

extern "C" void kernel_launch(void* const* d_in, const int* in_sizes, int n_in,
                              void* d_out, int out_size, void* d_ws, size_t ws_size,
                              hipStream_t stream) {
    // setup_inputs order:
    // 0 proposal_feat (unused), 1 target_candidate (unused), 2 candidate,
    // 3 RCNN_cls_result, 4 offset, 5 yaw_pred (unused), 6 y,
    // 7 y_yaw (unused), 8 horizon (unused)
    const float* candidate = (const float*)d_in[2];
    const float* rcnn      = (const float*)d_in[3];
    const float* offset    = (const float*)d_in[4];
    const float* y         = (const float*)d_in[6];

    float* out = (float*)d_out;
    float* ws  = (float*)d_ws;
    float* gt       = ws;        // 256 floats: (gx,gy) per batch
    float* partials = ws + 256;  // LOSS_GRID floats

    const int n4 = (B_SZ * N_SZ * 2) / 4;  // 4,194,304 float4s

    // 1) 64MB passthrough copy (out[0 .. B*N*2))
    copy_rcnn_kernel<<<2048, 256, 0, stream>>>((const f4*)rcnn, (f4*)out, n4);
    // 2) gt_last per batch
    gt_kernel<<<1, 128, 0, stream>>>(y, gt);
    // 3) distance-loss partials (argsort is permutation-invariant -> skipped)
    loss_partials_kernel<<<LOSS_GRID, 256, 0, stream>>>(
        (const f4*)candidate, (const f4*)offset, gt, partials, n4);
    // 4) deterministic final sum -> out[B*N*2]
    final_reduce_kernel<<<1, 256, 0, stream>>>(
        partials, LOSS_GRID, out + (size_t)B_SZ * N_SZ * 2);
}